// YOLOv3Loss_75642964017411
// MI455X (gfx1250) — compile-verified
//
#include <hip/hip_runtime.h>

// YOLOv3 loss for MI455X (gfx1250, wave32).
// Key insight: only channels 0..4 are needed densely (obj + ignore-mask IoU);
// xy/wh/cls terms are non-zero only at <=800 matched cells/layer (masked by
// tgt_mask/tgt_scale). So we read ~7MB instead of ~118MB.
// Wave reductions use V_WMMA_F32_16X16X4_F32 (ones-matrix trick).

#define NCLS 80
#define NCH  85
#define BB   16   // batch
#define TT   50   // truths per image
#define NA   3    // anchors per layer

typedef __attribute__((ext_vector_type(2))) float v2f;
typedef __attribute__((ext_vector_type(8))) float v8f;

__constant__ float c_anchw[9] = {10.f,16.f,33.f,30.f,62.f,59.f,116.f,156.f,373.f};
__constant__ float c_anchh[9] = {13.f,30.f,23.f,61.f,45.f,119.f,90.f,198.f,326.f};

__device__ __forceinline__ float bce(float x, float t) {
  // numerically stable BCE-with-logits
  return fmaxf(x, 0.0f) - x * t + log1pf(expf(-fabsf(x)));
}

// ---- wave32 sum via WMMA: A(16x4)=lane partials, B(4x16)=ones, D rows = s[m]+s[m+16].
// Lane 0 holds D rows 0..7, lane 16 rows 8..15 -> total = sum(d)/lane0 + sum(d)/lane16.
// Requires EXEC all-ones (callers keep all lanes active; inactive work contributes 0).
__device__ __forceinline__ float wave_sum_wmma(float v) {
  v2f a; a[0] = v;   a[1] = 0.0f;
  v2f b; b[0] = 1.f; b[1] = 1.0f;
  v8f c = {};
  v8f d = __builtin_amdgcn_wmma_f32_16x16x4_f32(false, a, false, b, (short)0, c,
                                                false, false);
  float r = ((d[0]+d[1])+(d[2]+d[3])) + ((d[4]+d[5])+(d[6]+d[7]));
  return __shfl(r, 0, 32) + __shfl(r, 16, 32);
}

__device__ __forceinline__ void block_atomic_add(float v, float* dst) {
  __shared__ float wsum[8];
  float w = wave_sum_wmma(v);
  int lane = threadIdx.x & 31;
  int wid  = threadIdx.x >> 5;
  int nw   = (blockDim.x + 31) >> 5;
  if (lane == 0) wsum[wid] = w;
  __syncthreads();
  if (threadIdx.x == 0) {
    float s = 0.f;
    for (int i = 0; i < nw; ++i) s += wsum[i];
    atomicAdd(dst, s);
  }
}

// ---------------- kernel 1: init loss scalar + owner arrays ----------------
__global__ void yolo_init_kernel(int* __restrict__ owner, int n, float* __restrict__ loss) {
  int i = blockIdx.x * blockDim.x + threadIdx.x;
  if (i == 0) loss[0] = 0.0f;
  for (; i < n; i += gridDim.x * blockDim.x) owner[i] = -1;
}

// ---------------- kernel 2: per-truth anchor matching (3 layers x B x T) ---
__global__ void yolo_prep_kernel(const float* __restrict__ targets,
                                 float* __restrict__ tF, int* __restrict__ tI,
                                 int* __restrict__ owner) {
  int gid = blockIdx.x * blockDim.x + threadIdx.x;
  if (gid >= 3 * BB * TT) return;
  int l = gid / (BB * TT);
  int r = gid % (BB * TT);
  int b = r / TT, t = r % TT;

  const int   dims[3]  = {19, 38, 76};
  const float strid[3] = {32.f, 16.f, 8.f};
  const int   abase[3] = {6, 3, 0};
  int ooff = 0;
  for (int k = 0; k < l; ++k) ooff += BB * NA * dims[k] * dims[k];

  int H = dims[l], W = H;
  float st = strid[l];

  const float* lab = targets + (size_t)(b * TT + t) * 5;
  float lc = lab[0], lx = lab[1], ly = lab[2], lw = lab[3], lh = lab[4];
  bool valid = (lc + lx + ly + lw + lh) > 0.0f;

  float tx = lx * W, ty = ly * H, tw = lw * W, th = lh * H;
  float area_t = tw * th;

  int best = 0; float bi = -1.0f;
  for (int k = 0; k < 9; ++k) {           // first-max wins, matches argmax
    float aw = c_anchw[k] / st, ah = c_anchh[k] / st;
    float iw = fminf(tw, aw), ih = fminf(th, ah);
    float inter = (iw > 0.f && ih > 0.f) ? iw * ih : 0.f;
    float iou = inter / (area_t + aw * ah - inter);
    if (iou > bi) { bi = iou; best = k; }
  }
  int bn = best % 3;
  int ab = abase[l];
  bool matched = valid && (best >= ab) && (best < ab + 3);

  float maw = c_anchw[ab + bn] / st, mah = c_anchh[ab + bn] / st;
  int ti = (int)tx, tj = (int)ty;
  float scale = sqrtf(2.0f - area_t / (float)(H * W));
  float t0 = tx - (float)ti, t1 = ty - (float)tj;
  float t2 = logf(tw / maw + 1e-16f), t3 = logf(th / mah + 1e-16f);

  float* rf = tF + (size_t)gid * 12;
  rf[0] = valid ? 1.f : 0.f;
  rf[1] = tx; rf[2] = ty; rf[3] = tw; rf[4] = th;
  rf[5] = scale; rf[6] = t0; rf[7] = t1; rf[8] = t2; rf[9] = t3;
  int* ri = tI + (size_t)gid * 8;
  ri[0] = matched ? 1 : 0; ri[1] = bn; ri[2] = ti; ri[3] = tj; ri[4] = (int)lc;

  if (matched && ti >= 0 && ti < W && tj >= 0 && tj < H)
    atomicMax(&owner[ooff + ((b * NA + bn) * H + tj) * W + ti], b * TT + t);
}

// ---------------- kernel 3: dense obj loss (channels 0..4 only) ------------
__global__ void yolo_obj_kernel(const float* __restrict__ out, const float* __restrict__ tF,
                                const int* __restrict__ owner, float* __restrict__ loss,
                                int H, int W, int layer, int abase, float stride) {
  __shared__ float tf[TT * 5];
  int b = blockIdx.y;
  for (int i = threadIdx.x; i < TT * 5; i += blockDim.x) {
    int t = i / 5, f = i % 5;
    tf[i] = tF[(size_t)((layer * BB + b) * TT + t) * 12 + f];
  }
  __syncthreads();

  float maw[3], mah[3];
  for (int k = 0; k < 3; ++k) {
    maw[k] = c_anchw[abase + k] / stride;
    mah[k] = c_anchh[abase + k] / stride;
  }

  int HW = H * W;
  int nc = NA * HW;
  int idx = blockIdx.x * blockDim.x + threadIdx.x;
  float part = 0.0f;
  if (idx < nc) {                 // guard sets part=0; all lanes stay active for WMMA
    int a = idx / HW, rem = idx % HW;
    int i = rem / W, j = rem % W;
    size_t base = ((size_t)(b * 255 + a * NCH) * H + i) * W + j;
    float x0 = out[base];
    float x1 = out[base + (size_t)HW];
    float x2 = out[base + 2 * (size_t)HW];
    float x3 = out[base + 3 * (size_t)HW];
    float x4 = out[base + 4 * (size_t)HW];

    float px = 1.0f / (1.0f + expf(-x0)) + (float)j;
    float py = 1.0f / (1.0f + expf(-x1)) + (float)i;
    float pw = expf(x2) * maw[a];
    float ph = expf(x3) * mah[a];
    float parea = pw * ph;

    float mi = 0.0f;
    for (int t = 0; t < TT; ++t) {
      const float* r = &tf[t * 5];
      if (r[0] > 0.0f) {
        float tx = r[1], ty = r[2], tw = r[3], th = r[4];
        float tlx = fmaxf(px - pw * 0.5f, tx - tw * 0.5f);
        float tly = fmaxf(py - ph * 0.5f, ty - th * 0.5f);
        float brx = fminf(px + pw * 0.5f, tx + tw * 0.5f);
        float bry = fminf(py + ph * 0.5f, ty + th * 0.5f);
        float wx = brx - tlx, wy = bry - tly;
        float inter = (wx > 0.f && wy > 0.f) ? wx * wy : 0.f;
        float iou = inter / (parea + tw * th - inter);
        mi = fmaxf(mi, iou);
      }
    }
    bool m = owner[(size_t)(b * NA + a) * HW + rem] >= 0;
    if (m)               part = bce(x4, 1.0f);          // matched cell: target obj = 1
    else if (mi <= 0.7f) part = bce(x4, 0.0f);          // kept cell
    else                 part = 0.69314718055994531f;   // ignored: bce(0,0) = ln 2
  }
  block_atomic_add(part, loss);
}

// ---------------- kernel 4: sparse matched-cell xy/wh/cls terms ------------
__global__ void yolo_matched_kernel(const float* __restrict__ out, const float* __restrict__ tF,
                                    const int* __restrict__ tI, const int* __restrict__ owner,
                                    float* __restrict__ loss, int H, int W, int layer) {
  int n = blockIdx.x;                 // = b*TT + t
  int b = n / TT;
  size_t rec = (size_t)layer * BB * TT + n;
  const int* ri = tI + rec * 8;
  if (!ri[0]) return;                 // block-uniform: EXEC stays all-ones below
  int a = ri[1], ti = ri[2], tj = ri[3], cls = ri[4];
  if (ti < 0 || ti >= W || tj < 0 || tj >= H) return;
  int HW = H * W;
  if (owner[(size_t)(b * NA + a) * HW + tj * W + ti] != n) return;  // duplicate lost

  const float* rf = tF + rec * 12;
  float scale = rf[5];
  float s2 = scale * scale;
  float tv0 = rf[6], tv1 = rf[7], tv2 = rf[8], tv3 = rf[9];

  float part = 0.0f;
  int c = threadIdx.x;
  if (c < NCH) {
    float x = out[((size_t)(b * 255 + a * NCH + c) * H + tj) * W + ti];
    if (c == 0)      part = bce(x, tv0) * s2;
    else if (c == 1) part = bce(x, tv1) * s2;
    else if (c == 2) { float d = x - tv2; part = 0.5f * s2 * d * d; }
    else if (c == 3) { float d = x - tv3; part = 0.5f * s2 * d * d; }
    else if (c == 4) part = 0.0f;     // obj handled densely
    else             part = bce(x, (c - 5 == cls) ? 1.0f : 0.0f);
  }
  block_atomic_add(part, loss);
}

extern "C" void kernel_launch(void* const* d_in, const int* in_sizes, int n_in,
                              void* d_out, int out_size, void* d_ws, size_t ws_size,
                              hipStream_t stream) {
  (void)in_sizes; (void)n_in; (void)out_size; (void)ws_size;
  const float* outs[3] = {(const float*)d_in[0], (const float*)d_in[1], (const float*)d_in[2]};
  const float* targets = (const float*)d_in[3];
  float* loss = (float*)d_out;

  // Workspace layout (~1.65 MB total):
  //   tF: 3*B*T records x 12 floats  (valid,tx,ty,tw,th,scale,t0..t3)
  //   tI: 3*B*T records x 8  ints    (matched,a,ti,tj,cls)
  //   owner: per-layer per-cell winning truth index (or -1)
  float* tF = (float*)d_ws;
  int*   tI = (int*)(tF + (size_t)3 * BB * TT * 12);
  int*   owner = tI + (size_t)3 * BB * TT * 8;

  const int   dims[3]    = {19, 38, 76};
  const float strides[3] = {32.f, 16.f, 8.f};
  const int   abase[3]   = {6, 3, 0};
  int ooff[3]; int tot = 0;
  for (int l = 0; l < 3; ++l) { ooff[l] = tot; tot += BB * NA * dims[l] * dims[l]; }

  yolo_init_kernel<<<256, 256, 0, stream>>>(owner, tot, loss);
  yolo_prep_kernel<<<(3 * BB * TT + 255) / 256, 256, 0, stream>>>(targets, tF, tI, owner);

  for (int l = 0; l < 3; ++l) {
    int H = dims[l], W = H;
    int nc = NA * H * W;
    dim3 g((nc + 255) / 256, BB);
    yolo_obj_kernel<<<g, 256, 0, stream>>>(outs[l], tF, owner + ooff[l], loss,
                                           H, W, l, abase[l], strides[l]);
    yolo_matched_kernel<<<BB * TT, 96, 0, stream>>>(outs[l], tF, tI, owner + ooff[l],
                                                    loss, H, W, l);
  }
}